// TrigHashGrid_60155311948498
// MI455X (gfx1250) — compile-verified
//
#include <hip/hip_runtime.h>

typedef float v2f __attribute__((ext_vector_type(2)));
typedef float v8f __attribute__((ext_vector_type(8)));

#define NLVL   16
#define MDIM   3
#define CCH    2
#define WGRID  1000
#define B_TOTAL 1048576
#define A_CUBIC (-0.75f)

__device__ __forceinline__ float cubic_inner(float s) {
    return ((A_CUBIC + 2.0f) * s - (A_CUBIC + 3.0f)) * s * s + 1.0f;
}
__device__ __forceinline__ float cubic_outer(float s) {
    return ((A_CUBIC * s - 5.0f * A_CUBIC) * s + 8.0f * A_CUBIC) * s - 4.0f * A_CUBIC;
}

__launch_bounds__(256, 2)
__global__ void trig_hashgrid_kernel(const float* __restrict__ x,
                                     const float* __restrict__ grids,
                                     const float* __restrict__ G,
                                     const float* __restrict__ H,
                                     float* __restrict__ out) {
    // grids (16,2,1000) f32 = 128 KB -> fits in CDNA5's 320 KB LDS per WGP
    __shared__ float sgrids[NLVL * CCH * WGRID];
    for (int i = threadIdx.x; i < NLVL * CCH * WGRID; i += blockDim.x)
        sgrids[i] = grids[i];
    __syncthreads();

    const int lane   = threadIdx.x & 31;   // wave32
    const int wave   = threadIdx.x >> 5;
    const int n      = lane & 15;          // level index this lane owns
    const bool hiHalf = (lane >= 16);

    // B matrices (K=4 x N=16, K=3 zero-padded) for the three m-slices.
    // G flat layout: G[k][m][n] at (k*3 + m)*16 + n.
    // Lanes 0-15 hold K=0,1 for N=lane; lanes 16-31 hold K=2,3 for N=lane-16.
    v2f bmat[MDIM];
#pragma unroll
    for (int m = 0; m < MDIM; ++m) {
        if (!hiHalf) {
            bmat[m][0] = G[(0 * MDIM + m) * NLVL + n];
            bmat[m][1] = G[(1 * MDIM + m) * NLVL + n];
        } else {
            bmat[m][0] = G[(2 * MDIM + m) * NLVL + n];
            bmat[m][1] = 0.0f;   // K=3 pad
        }
    }
    const float h0 = H[0 * NLVL + n];
    const float h1 = H[1 * NLVL + n];
    const float h2 = H[2 * NLVL + n];

    const int nTiles     = B_TOTAL / 16;
    const int wavesTotal = gridDim.x * (blockDim.x >> 5);
    const int waveGlobal = blockIdx.x * (blockDim.x >> 5) + wave;

    float2* __restrict__ out2 = (float2*)out;

    for (int tile = waveGlobal; tile < nTiles; tile += wavesTotal) {
        const int rowBase = tile * 16;

        // A matrix (M=16 x K=4, K=3 zero-padded):
        // lanes 0-15: (x[r][0], x[r][1]) for M=lane; lanes 16-31: (x[r][2], 0) for M=lane-16
        const int arow = rowBase + n;
        v2f amat;
        if (!hiHalf) {
            amat[0] = x[arow * 3 + 0];
            amat[1] = x[arow * 3 + 1];
        } else {
            amat[0] = x[arow * 3 + 2];
            amat[1] = 0.0f;
        }

        v8f acc0 = {}, acc1 = {}, acc2 = {};
        acc0 = __builtin_amdgcn_wmma_f32_16x16x4_f32(false, amat, false, bmat[0],
                                                     (short)0, acc0, false, false);
        acc1 = __builtin_amdgcn_wmma_f32_16x16x4_f32(false, amat, false, bmat[1],
                                                     (short)0, acc1, false, false);
        acc2 = __builtin_amdgcn_wmma_f32_16x16x4_f32(false, amat, false, bmat[2],
                                                     (short)0, acc2, false, false);

        // D layout: VGPR r holds (M=r, N=lane) for lanes 0-15, (M=r+8, N=lane-16) for 16-31
#pragma unroll
        for (int r = 0; r < 8; ++r) {
            const int brow = rowBase + r + (hiHalf ? 8 : 0);

            float g = __sinf(acc0[r] + h0) * __sinf(acc1[r] + h1) * __sinf(acc2[r] + h2);

            float ixf = ((g + 1.0f) * (float)WGRID - 1.0f) * 0.5f;
            float x0f = floorf(ixf);
            float t   = ixf - x0f;
            int  base = (int)x0f;

            float wk[4];
            wk[0] = cubic_outer(t + 1.0f);
            wk[1] = cubic_inner(t);
            wk[2] = cubic_inner(1.0f - t);
            wk[3] = cubic_outer(2.0f - t);

            float o0 = 0.0f, o1 = 0.0f;
            const int lvlBase = n * (CCH * WGRID);
#pragma unroll
            for (int k = 0; k < 4; ++k) {
                int  idx   = base + (k - 1);
                bool valid = (idx >= 0) && (idx < WGRID);
                int  idxc  = min(max(idx, 0), WGRID - 1);
                float wv   = valid ? wk[k] : 0.0f;
                o0 += sgrids[lvlBase + idxc] * wv;
                o1 += sgrids[lvlBase + WGRID + idxc] * wv;
            }
            // out (B, 16, 2): element (brow, n, c) -> coalesced float2 per lane
            out2[brow * NLVL + n] = make_float2(o0, o1);
        }
    }
}

extern "C" void kernel_launch(void* const* d_in, const int* in_sizes, int n_in,
                              void* d_out, int out_size, void* d_ws, size_t ws_size,
                              hipStream_t stream) {
    const float* x     = (const float*)d_in[0];
    const float* grids = (const float*)d_in[1];
    const float* G     = (const float*)d_in[2];
    const float* H     = (const float*)d_in[3];
    float* out = (float*)d_out;

    // 512 blocks x 8 waves = 4096 waves; 65536 tiles -> 16 tiles/wave,
    // amortizing the 128 KB LDS fill of `grids` per block.
    trig_hashgrid_kernel<<<dim3(512), dim3(256), 0, stream>>>(x, grids, G, H, out);
}